// ResGCN_82360292868394
// MI455X (gfx1250) — compile-verified
//
#include <hip/hip_runtime.h>
#include <hip/hip_bf16.h>

typedef __attribute__((ext_vector_type(2))) float v2f;
typedef __attribute__((ext_vector_type(4))) float v4f;
typedef __attribute__((ext_vector_type(8))) float v8f;

#define D 256            // D_IN == D_OUT == 256
#define LDS_STRIDE 260   // padded row stride (floats) -> bank-conflict-free B reads
#define LDS_BYTES (D * LDS_STRIDE * 4)  // 266240 B <= 320 KB WGP LDS

// ---------------------------------------------------------------------------
// Kernel A: W_eff = (S + I) * W, b_eff = (S + I) * b
// S[dst,src] += 1 for each edge (edge_mapping indexes the channel axis).
// One block per output row o; 256 threads = columns.
// ---------------------------------------------------------------------------
__global__ __launch_bounds__(256) void resgcn_build_weff(
    const float* __restrict__ W, const float* __restrict__ b,
    const long long* __restrict__ em, int E,
    float* __restrict__ Weff, float* __restrict__ beff)
{
    __shared__ int s0[1024];
    __shared__ int s1[1024];
    const int o = blockIdx.x;
    const int i = threadIdx.x;

    float acc  = W[o * D + i];           // identity term
    float bacc = (i == 0) ? b[o] : 0.0f;

    for (int base = 0; base < E; base += 1024) {
        const int chunk = min(1024, E - base);
        for (int e = threadIdx.x; e < chunk; e += blockDim.x) {
            s0[e] = (int)em[base + e];        // dst channel
            s1[e] = (int)em[E + base + e];    // src channel
        }
        __syncthreads();
        for (int e = 0; e < chunk; ++e) {
            if (s0[e] == o) {
                acc += W[s1[e] * D + i];
                if (i == 0) bacc += b[s1[e]];
            }
        }
        __syncthreads();
    }

    Weff[o * D + i] = acc;
    if (i == 0) beff[o] = bacc;
}

// ---------------------------------------------------------------------------
// Kernel B: out[m,o] = (sum_i X[m,i] * Weff[o,i] + beff[o]) / bl[m]
// f32 WMMA (16x16x4). Block = 256 threads = 8 waves, covers 128 rows x 256 cols.
// Wave layout: 4 row-groups x 2 col-groups; each wave: 32 rows x 128 cols
//   -> acc[2][8] 16x16 f32 tiles, K-loop of 64 steps (K=4 each).
// Weff staged once per block into padded LDS (B operand, reused by all waves).
// B fragments for each k-step are loaded into a distinct-register array first
// so the DS loads batch up and drain against the WMMA chain instead of
// serializing with per-load s_wait_dscnt 0.
// ---------------------------------------------------------------------------
__global__ __launch_bounds__(256) void resgcn_gemm_wmma(
    const float* __restrict__ X, const float* __restrict__ Weff,
    const float* __restrict__ beff, const float* __restrict__ bl,
    float* __restrict__ out, int M)
{
    extern __shared__ float lds[];

    const int tid = threadIdx.x;

    // ---- stage Weff (256x256 f32) into LDS with padded rows, b128 chunks ----
    for (int idx = tid; idx < D * (D / 4); idx += 256) {
        const int r  = idx >> 6;          // row 0..255
        const int c4 = (idx & 63) << 2;   // col 0..252 step 4
        const v4f v = *(const v4f*)(Weff + (size_t)r * D + c4);
        *(v4f*)(lds + r * LDS_STRIDE + c4) = v;   // 16B-aligned (1040 % 16 == 0)
    }
    __syncthreads();

    const int lane = tid & 31;
    const int w    = tid >> 5;          // wave 0..7
    const int rg   = w >> 1;            // row group 0..3
    const int cg   = w & 1;             // col group 0..1
    const int l16  = lane & 15;
    const int hl   = lane >> 4;         // 0 for lanes 0-15, 1 for lanes 16-31

    const int waveRow = blockIdx.x * 128 + rg * 32;
    const int waveCol = cg * 128;

    // A operand rows (clamped so OOB lanes read a valid address; results masked at store)
    int r0 = waveRow + l16;       if (r0 >= M) r0 = M - 1;
    int r1 = waveRow + 16 + l16;  if (r1 >= M) r1 = M - 1;

    // lane's A base: row start + column (2*hl); k-step advances 4 floats
    const float* a0base = X + (size_t)r0 * D + 2 * hl;
    const float* a1base = X + (size_t)r1 * D + 2 * hl;

    // per-n B base addresses in LDS (k-step advances 4 floats = 16 B)
    const float* bbase[8];
#pragma unroll
    for (int n = 0; n < 8; ++n)
        bbase[n] = lds + (waveCol + n * 16 + l16) * LDS_STRIDE + 2 * hl;

    v8f acc[2][8] = {};   // zero-initialized accumulators

    for (int kk = 0; kk < 64; ++kk) {
        const v2f a0 = *(const v2f*)(a0base + 4 * kk);
        const v2f a1 = *(const v2f*)(a1base + 4 * kk);

        v2f bb[8];
#pragma unroll
        for (int n = 0; n < 8; ++n)
            bb[n] = *(const v2f*)(bbase[n] + 4 * kk);

#pragma unroll
        for (int n = 0; n < 8; ++n) {
            acc[0][n] = __builtin_amdgcn_wmma_f32_16x16x4_f32(
                false, a0, false, bb[n], (short)0, acc[0][n], false, false);
            acc[1][n] = __builtin_amdgcn_wmma_f32_16x16x4_f32(
                false, a1, false, bb[n], (short)0, acc[1][n], false, false);
        }
    }

    // ---- epilogue: add b_eff, divide by batch_lens, predicated store ----
#pragma unroll
    for (int t = 0; t < 2; ++t) {
#pragma unroll
        for (int n = 0; n < 8; ++n) {
            const int col = waveCol + n * 16 + l16;
            const float be = beff[col];
#pragma unroll
            for (int r = 0; r < 8; ++r) {
                // D layout: VGPR r -> M = r (lanes 0-15) / M = 8 + r (lanes 16-31)
                const int row = waveRow + t * 16 + hl * 8 + r;
                if (row < M) {
                    const float d = bl[row];
                    out[(size_t)row * D + col] = (acc[t][n][r] + be) / d;
                }
            }
        }
    }
}

// ---------------------------------------------------------------------------
extern "C" void kernel_launch(void* const* d_in, const int* in_sizes, int n_in,
                              void* d_out, int out_size, void* d_ws, size_t ws_size,
                              hipStream_t stream)
{
    const float*     X  = (const float*)d_in[0];      // [1, M, 256] f32
    const float*     W  = (const float*)d_in[1];      // [256, 256]  f32
    const float*     b  = (const float*)d_in[2];      // [256]       f32
    const long long* em = (const long long*)d_in[3];  // [2, E]      int64
    const float*     bl = (const float*)d_in[4];      // [M, 1]      f32
    float*           out = (float*)d_out;             // [1, M, 256] f32

    const int M = in_sizes[0] / D;
    const int E = in_sizes[3] / 2;

    float* Weff = (float*)d_ws;          // 256*256 f32
    float* beff = Weff + D * D;          // 256 f32

    resgcn_build_weff<<<D, 256, 0, stream>>>(W, b, em, E, Weff, beff);

    const int grid = (M + 127) / 128;
    resgcn_gemm_wmma<<<grid, 256, LDS_BYTES, stream>>>(X, Weff, beff, bl, out, M);
}